// DynamicKSelector_49185965474231
// MI455X (gfx1250) — compile-verified
//
#include <hip/hip_runtime.h>
#include <hip/hip_bf16.h>

// Reference shapes: R_combined [16, 4096, 1024] f32, k_logits [3,5] f32,
// temperature scalar f32. Outputs concatenated flat:
//   k_selected[3] | R_star_levels[3,16,5,1024] | R_idx_levels[3,16,5,1024]
//
// Block layout: 512 blocks x 256 threads. Block b covers n = b>>5 and a
// 32-column hw stripe; thread (seg, c) scans rows [seg*512, seg*512+512) of
// column hw0+c. Each wave streams its rows as 16x32 f32 tiles staged by the
// CDNA5 Tensor Data Mover into a private 4-buffer LDS ring with 3 DMAs in
// flight (TENSORcnt-paced, no barriers in steady state). Pipeline depth is
// sized from the BW*latency product: 4096 waves * 3 * 2KB ~= 24 MB in flight
// vs ~16 MB needed to fill 23.3 TB/s at ~700 ns. Per-column partial top-5s
// are then merged through LDS and masked per level.

typedef unsigned int u32x4 __attribute__((ext_vector_type(4)));
typedef int          i32x4 __attribute__((ext_vector_type(4)));
typedef int          i32x8 __attribute__((ext_vector_type(8)));

#if defined(__has_builtin)
#if __has_builtin(__builtin_amdgcn_tensor_load_to_lds)
#define HAVE_TDM 1
#endif
#if __has_builtin(__builtin_amdgcn_s_wait_tensorcnt)
#define WAIT_TENSOR(n) __builtin_amdgcn_s_wait_tensorcnt(n)
#endif
#endif
#ifndef WAIT_TENSOR
#define WAIT_TENSOR(n)
#endif

#ifdef HAVE_TDM
// One TDM 2D tile load: 16 rows x 32 cols of f32, row stride 1024 elements.
// D# layout per the CDNA5 async-tensor chapter, section 8 (groups 0/1; 2D
// tensor so groups 2/3 are zero).
__device__ __forceinline__ void tdm_load_tile(const float* gaddr, unsigned int lds_addr) {
    const unsigned long long ga = (unsigned long long)gaddr;
    u32x4 g0;
    g0.x = 0x1u;                                                    // count=1, user desc
    g0.y = lds_addr;                                                // lds_addr [63:32]
    g0.z = (unsigned int)ga;                                        // global_addr[31:0]
    g0.w = (unsigned int)((ga >> 32) & 0x01FFFFFFu) | (2u << 30);   // ga[56:32] | type=2
    i32x8 g1;
    g1[0] = 0x00020000;       // workgroup_mask=0, data_size=2 (4 bytes)
    g1[1] = 1024 << 16;       // tensor_dim0[15:0] at bits [63:48]
    g1[2] = 4096 << 16;       // tensor_dim0 hi = 0; tensor_dim1[15:0] at bits [95:80]
    g1[3] = 32 << 16;         // tensor_dim1 hi = 0; tile_dim0 = 32 at bits [127:112]
    g1[4] = 16;               // tile_dim1 = 16, tile_dim2 = 0
    g1[5] = 1024;             // tensor_dim0_stride[31:0]
    g1[6] = 0;
    g1[7] = 0;
    i32x4 z4 = {0, 0, 0, 0};
#if __clang_major__ >= 23
    i32x8 z8 = {0, 0, 0, 0, 0, 0, 0, 0};
    __builtin_amdgcn_tensor_load_to_lds(g0, g1, z4, z4, z8, 0);   // 6-arg (clang-23)
#else
    __builtin_amdgcn_tensor_load_to_lds(g0, g1, z4, z4, 0);       // 5-arg (ROCm clang-22)
#endif
}
#endif

__device__ __forceinline__ unsigned int tf_rotl(unsigned int x, int d) {
    return (x << d) | (x >> (32 - d));
}

__global__ __launch_bounds__(256)
void DynamicKSelector_49185965474231_kernel(const float* R,
                                            const float* klog,
                                            const float* temp,
                                            float* out) {
    const int tid = threadIdx.x;
    const int n   = blockIdx.x >> 5;            // 0..15
    const int hw0 = (blockIdx.x & 31) * 32;     // column stripe base
    const int seg = tid >> 5;                   // 0..7 (one wave per segment)
    const int c   = tid & 31;                   // column within stripe
    const int hw  = hw0 + c;
    const float* base = R + (unsigned long long)n * 4096ull * 1024ull;

    __shared__ float tile[8][4][16 * 32];       // per-wave 4-buffer ring (64 KB)
    __shared__ float pv[256 * 5];               // per-thread partial top-5 values
    __shared__ int   pi[256 * 5];               // per-thread partial top-5 indices

    // ---- streaming sorted top-5 over rows [seg*512, seg*512+512) ----
    float t0 = -__builtin_inff(), t1 = t0, t2 = t0, t3 = t0, t4 = t0;
    int   i0 = 0, i1 = 0, i2 = 0, i3 = 0, i4 = 0;
    const int row0 = seg * 512;

#ifdef HAVE_TDM
    // Tile t covers rows row0 + t*16 .. +16 of the 32-column stripe (32 tiles).
    tdm_load_tile(base + (unsigned long long)(row0 +  0) * 1024ull + hw0,
                  (unsigned int)(size_t)(&tile[seg][0][0]));
    tdm_load_tile(base + (unsigned long long)(row0 + 16) * 1024ull + hw0,
                  (unsigned int)(size_t)(&tile[seg][1][0]));
    tdm_load_tile(base + (unsigned long long)(row0 + 32) * 1024ull + hw0,
                  (unsigned int)(size_t)(&tile[seg][2][0]));
    for (int t = 0; t < 32; ++t) {
        // TDM ops from one wave complete in order; keep up to 3 tiles in flight.
        if (t < 30)      { WAIT_TENSOR(2); }
        else if (t < 31) { WAIT_TENSOR(1); }
        else             { WAIT_TENSOR(0); }
        asm volatile("" ::: "memory");          // compiler fence only, no insts
        const int bufid = t & 3;
        const int hrb = row0 + t * 16;
        for (int r = 0; r < 16; ++r) {
            const float v = tile[seg][bufid][r * 32 + c];
            const int hr = hrb + r;
            if (v > t4) {
                const bool c0 = v > t0, c1 = v > t1, c2 = v > t2, c3 = v > t3;
                t4 = c3 ? t3 : v;              i4 = c3 ? i3 : hr;
                t3 = c2 ? t2 : (c3 ? v : t3);  i3 = c2 ? i2 : (c3 ? hr : i3);
                t2 = c1 ? t1 : (c2 ? v : t2);  i2 = c1 ? i1 : (c2 ? hr : i2);
                t1 = c0 ? t0 : (c1 ? v : t1);  i1 = c0 ? i0 : (c1 ? hr : i1);
                t0 = c0 ? v  : t0;             i0 = c0 ? hr : i0;
            }
        }
        asm volatile("" ::: "memory");
        if (t + 3 < 32)                          // refill the buffer drained at t-1
            tdm_load_tile(base + (unsigned long long)(row0 + (t + 3) * 16) * 1024ull + hw0,
                          (unsigned int)(size_t)(&tile[seg][(t + 3) & 3][0]));
    }
#else
    for (int r = 0; r < 512; ++r) {
        const int hr = row0 + r;
        const float v = base[(unsigned long long)hr * 1024ull + hw];
        if (v > t4) {
            const bool c0 = v > t0, c1 = v > t1, c2 = v > t2, c3 = v > t3;
            t4 = c3 ? t3 : v;              i4 = c3 ? i3 : hr;
            t3 = c2 ? t2 : (c3 ? v : t3);  i3 = c2 ? i2 : (c3 ? hr : i3);
            t2 = c1 ? t1 : (c2 ? v : t2);  i2 = c1 ? i1 : (c2 ? hr : i2);
            t1 = c0 ? t0 : (c1 ? v : t1);  i1 = c0 ? i0 : (c1 ? hr : i1);
            t0 = c0 ? v  : t0;             i0 = c0 ? hr : i0;
        }
    }
#endif

    // ---- publish per-(segment, column) partial top-5 to LDS ----
    pv[tid * 5 + 0] = t0;  pi[tid * 5 + 0] = i0;
    pv[tid * 5 + 1] = t1;  pi[tid * 5 + 1] = i1;
    pv[tid * 5 + 2] = t2;  pi[tid * 5 + 2] = i2;
    pv[tid * 5 + 3] = t3;  pi[tid * 5 + 3] = i3;
    pv[tid * 5 + 4] = t4;  pi[tid * 5 + 4] = i4;
    __syncthreads();

    if (tid < 32) {
        // ---- JAX threefry2x32 Gumbel-softmax hard k per level ----
        unsigned int bits[15];
        for (int j = 0; j < 8; ++j) {
            const unsigned int k0 = 0u, k1 = 42u;
            const unsigned int ks2 = 0x1BD11BDAu ^ k0 ^ k1;
            const int rotA[4] = {13, 15, 26, 6};
            const int rotB[4] = {17, 29, 16, 24};
            unsigned int x0 = (unsigned int)j + k0;
            unsigned int x1 = ((j < 7) ? (unsigned int)(8 + j) : 0u) + k1;
            for (int i = 0; i < 4; ++i) { x0 += x1; x1 = tf_rotl(x1, rotA[i]); x1 ^= x0; }
            x0 += k1;  x1 += ks2 + 1u;
            for (int i = 0; i < 4; ++i) { x0 += x1; x1 = tf_rotl(x1, rotB[i]); x1 ^= x0; }
            x0 += ks2; x1 += k0 + 2u;
            for (int i = 0; i < 4; ++i) { x0 += x1; x1 = tf_rotl(x1, rotA[i]); x1 ^= x0; }
            x0 += k0;  x1 += k1 + 3u;
            for (int i = 0; i < 4; ++i) { x0 += x1; x1 = tf_rotl(x1, rotB[i]); x1 ^= x0; }
            x0 += k1;  x1 += ks2 + 4u;
            for (int i = 0; i < 4; ++i) { x0 += x1; x1 = tf_rotl(x1, rotA[i]); x1 ^= x0; }
            x0 += ks2; x1 += k0 + 5u;
            bits[j] = x0;
            if (j < 7) bits[8 + j] = x1;
        }
        const float T = temp[0];
        int khard[3];
        for (int l = 0; l < 3; ++l) {
            int kh = 1;
            float best = -__builtin_inff();
            for (int r = 0; r < 5; ++r) {
                const unsigned int bb = bits[l * 5 + r];
                float f = __uint_as_float((bb >> 9) | 0x3f800000u) - 1.0f;
                float u = fmaxf(1e-6f, 1e-6f + f * (1.0f - 2e-6f));
                float g = -logf(-logf(u));
                float z = (klog[l * 5 + r] + g) / T;
                if (z > best) { best = z; kh = r + 1; }
            }
            khard[l] = kh;
        }
        if (blockIdx.x == 0 && tid == 0) {
            out[0] = (float)khard[0];
            out[1] = (float)khard[1];
            out[2] = (float)khard[2];
        }

        // ---- merge 8 segment partials for column c = tid ----
        // scan order (seg asc, rank asc) == ascending hr blocks, so strict >
        // keeps the lowest hr among equal values (jax.lax.top_k tie order).
        float mv[40]; int mi[40];
        for (int s = 0; s < 8; ++s) {
            for (int r = 0; r < 5; ++r) {
                mv[s * 5 + r] = pv[(s * 32 + tid) * 5 + r];
                mi[s * 5 + r] = pi[(s * 32 + tid) * 5 + r];
            }
        }
        float tv[5]; int ti[5];
        for (int s = 0; s < 5; ++s) {
            float bv = -__builtin_inff();
            int bi = 0, bj = 0;
            for (int j = 0; j < 40; ++j) {
                if (mv[j] > bv) { bv = mv[j]; bi = mi[j]; bj = j; }
            }
            tv[s] = bv; ti[s] = bi;
            for (int j = 0; j < 40; ++j) mv[j] = (j == bj) ? -__builtin_inff() : mv[j];
        }

        // ---- per-level masked outputs (coalesced across the 32 lanes) ----
        for (int l = 0; l < 3; ++l) {
            const int kh = khard[l];
            for (int r = 0; r < 5; ++r) {
                const int o = ((l * 16 + n) * 5 + r) * 1024 + hw0 + tid;
                const bool m = r < kh;
                out[3 + o]          = m ? tv[r] : -__builtin_inff();
                out[3 + 245760 + o] = m ? (float)ti[r] : 0.0f;
            }
        }
    }
}

extern "C" void kernel_launch(void* const* d_in, const int* in_sizes, int n_in,
                              void* d_out, int out_size, void* d_ws, size_t ws_size,
                              hipStream_t stream) {
    const float* R    = (const float*)d_in[0];
    const float* klog = (const float*)d_in[1];
    const float* temp = (const float*)d_in[2];
    (void)in_sizes; (void)n_in; (void)out_size; (void)d_ws; (void)ws_size;

    DynamicKSelector_49185965474231_kernel<<<512, 256, 0, stream>>>(
        R, klog, temp, (float*)d_out);
}